// MMHG_30743375905446
// MI455X (gfx1250) — compile-verified
//
#include <hip/hip_runtime.h>
#include <cstddef>

typedef __attribute__((ext_vector_type(2))) float v2f;
typedef __attribute__((ext_vector_type(8))) float v8f;

#define B_SZ   32
#define L_SEQ  300
#define D_EMB  256
#define N_TOK  (B_SZ * L_SEQ)      // 9600
#define NNZ_C  100000
#define E_C    4000
#define DF_    384
#define DU_    128

// 32-bit LDS byte address of a generic pointer to __shared__ memory
static __device__ __forceinline__ unsigned lds_addr_of(const void* p) {
  return (unsigned)(unsigned long long)(const __attribute__((address_space(3))) void*)p;
}

// -------------------------------------------------------------------------
// Tiled fp32 WMMA GEMM:  out(M x Ndim) = gather(Xtab)(M x Kdim) @ W(Kdim x Ndim)
// mode: 0 = plain, 1 = +bias, 2 = +bias + 0.002*sinusoidal-PE (Ndim==256),
//       3 = +bias + relu
// Block: 256 threads (8 waves). Each block: 16 rows x 128 cols.
// Wave w owns the 16x16 tile at cols nblock + w*16.
// A and W tiles staged with GLOBAL_LOAD_ASYNC_TO_LDS_B128 (ASYNCcnt path);
// inner loop is ds_load + v_wmma only.
// Fragment layouts per CDNA5 ISA 7.12.2 (f32 16x16x4).
// -------------------------------------------------------------------------
__global__ __launch_bounds__(256) void wmma_gemm(
    const float* __restrict__ Xtab, const int* __restrict__ gidx,
    const float* __restrict__ W, const float* __restrict__ bias,
    float* __restrict__ out, int Kdim, int Ndim, int mode)
{
  constexpr int KC  = 64;
  constexpr int LDA = KC + 2;      // 66: A reads hit distinct banks (2*mm spread)
  constexpr int LDW = 136;         // 128 + 8: K+2 row lands 16 banks away
  __shared__ __align__(16) float As[16 * LDA];
  __shared__ __align__(16) float Ws[KC * LDW];

  const int tid    = threadIdx.x;
  const int wave   = tid >> 5;
  const int lane   = tid & 31;
  const int row0   = blockIdx.x << 4;
  const int nblock = blockIdx.y << 7;
  const int mm     = lane & 15;
  const int kk     = (lane >> 4) << 1;   // 0 for lanes 0-15, 2 for lanes 16-31

  v8f acc = {};

  // A-tile loader mapping: 16 rows x 64 cols, one b128 per thread
  const int  ldr = tid >> 4;
  const int  ldc = (tid & 15) << 2;
  const long srow = gidx ? (long)gidx[row0 + ldr] : (long)(row0 + ldr);
  const float* srcbase = Xtab + srow * (long)Kdim + ldc;
  const unsigned adst  = lds_addr_of(&As[ldr * LDA + ldc]);

  // W-tile loader mapping: 64 rows x 128 cols, 8 b128 per thread
  const float* wbase = W + nblock;

  for (int kc = 0; kc < Kdim; kc += KC) {
    // ---- stage A tile: async global -> LDS, no VGPR round-trip ----
    {
      const float* agp = srcbase + kc;
      asm volatile("global_load_async_to_lds_b128 %0, %1, off"
                   :: "v"(adst), "v"(agp) : "memory");
      if (kc + KC < Kdim)
        __builtin_prefetch(srcbase + kc + KC, 0, 0);   // global_prefetch_b8
    }
    // ---- stage W tile: 8 async b128 per thread, all in flight ----
#pragma unroll
    for (int i = 0; i < 8; ++i) {
      int f   = tid + (i << 8);
      int wr  = f >> 5;              // 0..63
      int wc4 = (f & 31) << 2;       // 0..124
      const unsigned wdst = lds_addr_of(&Ws[wr * LDW + wc4]);
      const float*   wgp  = wbase + (long)(kc + wr) * Ndim + wc4;
      asm volatile("global_load_async_to_lds_b128 %0, %1, off"
                   :: "v"(wdst), "v"(wgp) : "memory");
    }
    asm volatile("s_wait_asynccnt 0" ::: "memory");
    __syncthreads();

    // ---- 16 x WMMA, all operands from LDS ----
    const float* __restrict__ Ap = &As[mm * LDA + kk];
    const float* __restrict__ Bp = &Ws[kk * LDW + (wave << 4) + mm];
#pragma unroll
    for (int k = 0; k < KC; k += 4) {
      v2f a = *(const v2f*)(Ap + k);   // A frag: K = k+kk, k+kk+1
      v2f b;
      b.x = Bp[(long)k * LDW];         // B frag: rows k+kk, k+kk+1 at this col
      b.y = Bp[(long)(k + 1) * LDW];
      acc = __builtin_amdgcn_wmma_f32_16x16x4_f32(
          /*neg_a=*/false, a, /*neg_b=*/false, b,
          /*c_mod=*/(short)0, acc, /*reuse_a=*/false, /*reuse_b=*/false);
    }
    __syncthreads();
  }

  // Epilogue: C VGPR r -> row (r + 8*(lane>>4)), col = nblock + wave*16 + (lane&15)
  const int   col   = nblock + (wave << 4) + mm;
  const int   rbase = row0 + ((lane >> 4) << 3);
  const float bv    = (mode >= 1) ? bias[col] : 0.0f;
#pragma unroll
  for (int r = 0; r < 8; ++r) {
    int   m = rbase + r;
    float v = acc[r] + bv;
    if (mode == 2) {
      int l = m % L_SEQ;                    // position within sequence
      if (l > 0) {
        int   d2   = col >> 1;
        float freq = __expf((float)(2 * d2) * (-9.210340371976184f / (float)D_EMB));
        float ang  = (float)(l - 1) * freq;
        v += 0.002f * ((col & 1) ? __cosf(ang) : __sinf(ang));
      }
    }
    if (mode == 3) v = fmaxf(v, 0.0f);
    out[(long)m * Ndim + col] = v;
  }
}

// ---------------- hypergraph degree counts ----------------
__global__ void k_deg(const int* __restrict__ node_idx, const int* __restrict__ edge_idx,
                      float* __restrict__ deg_n, float* __restrict__ deg_e, int nnz)
{
  int i = blockIdx.x * blockDim.x + threadIdx.x;
  if (i < nnz) {
    atomicAdd(&deg_e[edge_idx[i]], 1.0f);
    atomicAdd(&deg_n[node_idx[i]], 1.0f);
  }
}

// dst[dIdx[nz]][:] += src[sIdx[nz]][:]  (rows of 256 floats, float4 atomics)
__global__ __launch_bounds__(256) void k_seg_scatter(
    const float* __restrict__ src, const int* __restrict__ sIdx,
    const int* __restrict__ dIdx, float* __restrict__ dst, int nnz)
{
  int t  = threadIdx.x;
  int nz = blockIdx.x * 4 + (t >> 6);
  if (nz >= nnz) return;
  int c = (t & 63) << 2;
  const float4 v = *(const float4*)(src + (long)sIdx[nz] * D_EMB + c);
  float* d = dst + (long)dIdx[nz] * D_EMB + c;
  atomicAdd(d + 0, v.x); atomicAdd(d + 1, v.y);
  atomicAdd(d + 2, v.z); atomicAdd(d + 3, v.w);
}

// x[row][col] = x[row][col] / max(deg[row],1) (+ bias[col])
__global__ void k_norm(float* __restrict__ x, const float* __restrict__ deg,
                       const float* __restrict__ bias, int rows)
{
  long i = (long)blockIdx.x * blockDim.x + threadIdx.x;
  if (i >= (long)rows * D_EMB) return;
  int r = (int)(i >> 8), c = (int)(i & 255);
  float v = x[i] / fmaxf(deg[r], 1.0f);
  if (bias) v += bias[c];
  x[i] = v;
}

// Build text_user / img_user / kv-stack / concat from position-0 tokens + user embeds
__global__ __launch_bounds__(DF_) void k_assemble(
    const float* __restrict__ tg, const float* __restrict__ ig,
    const float* __restrict__ user_table, const int* __restrict__ uid,
    float* __restrict__ kvcat, float* __restrict__ cat)
{
  int b = blockIdx.x, j = threadIdx.x;
  float tv, iv;
  if (j < D_EMB) {
    tv = tg[(long)b * L_SEQ * D_EMB + j];
    iv = ig[(long)b * L_SEQ * D_EMB + j];
  } else {
    float u = user_table[(long)uid[b] * DU_ + (j - D_EMB)];
    tv = u; iv = u;
  }
  kvcat[(long)b * DF_ + j]            = tv;   // kv[b,0] = text_user
  kvcat[(long)(B_SZ + b) * DF_ + j]   = iv;   // kv[b,1] = img_user
  cat[(long)b * 2 * DF_ + j]          = tv;
  cat[(long)b * 2 * DF_ + DF_ + j]    = iv;
}

// 4-head, 2-key softmax attention (dh = 96)
__global__ __launch_bounds__(DF_) void k_attn(
    const float* __restrict__ Q, const float* __restrict__ Kc,
    const float* __restrict__ Vc, float* __restrict__ aout)
{
  __shared__ float sS[8];
  __shared__ float sA[8];
  int b = blockIdx.x, t = threadIdx.x;
  if (t < 8) {
    int h = t >> 1, k = t & 1;
    const float* q  = Q  + (long)b * DF_ + h * 96;
    const float* kr = Kc + (long)(k * B_SZ + b) * DF_ + h * 96;
    float s = 0.0f;
    for (int i = 0; i < 96; ++i) s += q[i] * kr[i];
    sS[t] = s * 0.10206207261596575f;   // 1/sqrt(96)
  }
  __syncthreads();
  if (t < 4) {
    float s0 = sS[2 * t], s1 = sS[2 * t + 1];
    float m  = fmaxf(s0, s1);
    float e0 = __expf(s0 - m), e1 = __expf(s1 - m);
    float inv = 1.0f / (e0 + e1);
    sA[2 * t] = e0 * inv; sA[2 * t + 1] = e1 * inv;
  }
  __syncthreads();
  int h = t / 96;
  aout[(long)b * DF_ + t] = sA[2 * h]     * Vc[(long)b * DF_ + t]
                          + sA[2 * h + 1] * Vc[(long)(B_SZ + b) * DF_ + t];
}

__global__ void k_add(const float* __restrict__ a, const float* __restrict__ b,
                      float* __restrict__ c, int n)
{
  int i = blockIdx.x * blockDim.x + threadIdx.x;
  if (i < n) c[i] = a[i] + b[i];
}

// out[b] = dot(hidden[b], W4) + b4
__global__ __launch_bounds__(DF_) void k_final(
    const float* __restrict__ hidden, const float* __restrict__ W4,
    const float* __restrict__ b4, float* __restrict__ out)
{
  __shared__ float red[512];
  int b = blockIdx.x, t = threadIdx.x;
  red[t] = hidden[(long)b * DF_ + t] * W4[t];
  if (t < 128) red[DF_ + t] = 0.0f;
  __syncthreads();
  for (int s = 256; s > 0; s >>= 1) {
    if (t < s) red[t] += red[t + s];
    __syncthreads();
  }
  if (t == 0) out[b] = red[0] + b4[0];
}

// -------------------------------------------------------------------------
extern "C" void kernel_launch(void* const* d_in, const int* in_sizes, int n_in,
                              void* d_out, int out_size, void* d_ws, size_t ws_size,
                              hipStream_t stream)
{
  (void)in_sizes; (void)n_in; (void)out_size; (void)ws_size;
  const float* text_table = (const float*)d_in[2];
  const float* img_table  = (const float*)d_in[3];
  const float* W1  = (const float*)d_in[4];
  const float* b1  = (const float*)d_in[5];
  const float* W2  = (const float*)d_in[6];
  const float* b2  = (const float*)d_in[7];
  const float* user_table = (const float*)d_in[8];
  const float* Wt  = (const float*)d_in[9];
  const float* bt  = (const float*)d_in[10];
  const float* Wi  = (const float*)d_in[11];
  const float* bi  = (const float*)d_in[12];
  const float* Wq  = (const float*)d_in[13];
  const float* Wk  = (const float*)d_in[14];
  const float* Wv  = (const float*)d_in[15];
  const float* Wo  = (const float*)d_in[16];
  const float* W3  = (const float*)d_in[17];
  const float* b3  = (const float*)d_in[18];
  const float* Wd  = (const float*)d_in[19];
  const float* bd  = (const float*)d_in[20];
  const float* W4  = (const float*)d_in[21];
  const float* b4  = (const float*)d_in[22];
  const int* related  = (const int*)d_in[23];
  const int* uid      = (const int*)d_in[24];
  const int* node_idx = (const int*)d_in[25];
  const int* edge_idx = node_idx + NNZ_C;
  float* out = (float*)d_out;

  // ---- workspace carve-up (floats) ----
  const long ND = (long)N_TOK * D_EMB;        // 2,457,600
  float* p     = (float*)d_ws;
  float* tg    = p; p += ND;
  float* ig    = p; p += ND;
  float* tmp   = p; p += ND;
  float* ebuf  = p; p += (long)E_C * D_EMB;   // 1,024,000
  float* deg_e = p; p += 4096;
  float* deg_n = p; p += 9728;
  float* kvcat = p; p += 2 * B_SZ * DF_;
  float* cat   = p; p += B_SZ * 2 * DF_;
  float* fused = p; p += B_SZ * DF_;
  float* Qb    = p; p += B_SZ * DF_;
  float* Kcb   = p; p += 2 * B_SZ * DF_;
  float* Vcb   = p; p += 2 * B_SZ * DF_;
  float* aoutb = p; p += B_SZ * DF_;
  float* oOb   = p; p += B_SZ * DF_;
  float* obuf  = p; p += B_SZ * DF_;
  float* hid   = p; p += B_SZ * DF_;

  // ---- degrees (layer-invariant) ----
  hipMemsetAsync(deg_e, 0, E_C * sizeof(float), stream);
  hipMemsetAsync(deg_n, 0, N_TOK * sizeof(float), stream);
  k_deg<<<(NNZ_C + 255) / 256, 256, 0, stream>>>(node_idx, edge_idx, deg_n, deg_e, NNZ_C);

  // ---- embeddings (gather + GEMM + bias + PE fused) ----
  wmma_gemm<<<dim3(N_TOK / 16, 2), 256, 0, stream>>>(
      text_table, related, W1, b1, tg, 384, D_EMB, 2);
  wmma_gemm<<<dim3(N_TOK / 16, 2), 256, 0, stream>>>(
      img_table, related, W2, b2, ig, 2048, D_EMB, 2);

  // ---- 2 hypergraph conv layers x 2 modalities ----
  for (int l = 0; l < 2; ++l) {
    // text: tmp = hconv(tg, Wt[l], bt[l]); swap
    wmma_gemm<<<dim3(N_TOK / 16, 2), 256, 0, stream>>>(
        tg, nullptr, Wt + (long)l * D_EMB * D_EMB, nullptr, tmp, D_EMB, D_EMB, 0);
    hipMemsetAsync(ebuf, 0, (size_t)E_C * D_EMB * sizeof(float), stream);
    k_seg_scatter<<<NNZ_C / 4, 256, 0, stream>>>(tmp, node_idx, edge_idx, ebuf, NNZ_C);
    k_norm<<<(E_C * D_EMB) / 256, 256, 0, stream>>>(ebuf, deg_e, nullptr, E_C);
    hipMemsetAsync(tmp, 0, (size_t)ND * sizeof(float), stream);
    k_seg_scatter<<<NNZ_C / 4, 256, 0, stream>>>(ebuf, edge_idx, node_idx, tmp, NNZ_C);
    k_norm<<<(N_TOK * D_EMB) / 256, 256, 0, stream>>>(tmp, deg_n, bt + (long)l * D_EMB, N_TOK);
    { float* s = tg; tg = tmp; tmp = s; }

    // img: tmp = hconv(ig, Wi[l], bi[l]); swap
    wmma_gemm<<<dim3(N_TOK / 16, 2), 256, 0, stream>>>(
        ig, nullptr, Wi + (long)l * D_EMB * D_EMB, nullptr, tmp, D_EMB, D_EMB, 0);
    hipMemsetAsync(ebuf, 0, (size_t)E_C * D_EMB * sizeof(float), stream);
    k_seg_scatter<<<NNZ_C / 4, 256, 0, stream>>>(tmp, node_idx, edge_idx, ebuf, NNZ_C);
    k_norm<<<(E_C * D_EMB) / 256, 256, 0, stream>>>(ebuf, deg_e, nullptr, E_C);
    hipMemsetAsync(tmp, 0, (size_t)ND * sizeof(float), stream);
    k_seg_scatter<<<NNZ_C / 4, 256, 0, stream>>>(ebuf, edge_idx, node_idx, tmp, NNZ_C);
    k_norm<<<(N_TOK * D_EMB) / 256, 256, 0, stream>>>(tmp, deg_n, bi + (long)l * D_EMB, N_TOK);
    { float* s = ig; ig = tmp; tmp = s; }
  }

  // ---- fusion + attention head ----
  k_assemble<<<B_SZ, DF_, 0, stream>>>(tg, ig, user_table, uid, kvcat, cat);

  // fused = cat(32x768) @ W3 + b3
  wmma_gemm<<<dim3(2, 3), 256, 0, stream>>>(cat, nullptr, W3, b3, fused, 2 * DF_, DF_, 1);
  // Q = fused @ Wq ; K/V on stacked kv (64 rows)
  wmma_gemm<<<dim3(2, 3), 256, 0, stream>>>(fused, nullptr, Wq, nullptr, Qb, DF_, DF_, 0);
  wmma_gemm<<<dim3(4, 3), 256, 0, stream>>>(kvcat, nullptr, Wk, nullptr, Kcb, DF_, DF_, 0);
  wmma_gemm<<<dim3(4, 3), 256, 0, stream>>>(kvcat, nullptr, Wv, nullptr, Vcb, DF_, DF_, 0);

  k_attn<<<B_SZ, DF_, 0, stream>>>(Qb, Kcb, Vcb, aoutb);

  // o = attn @ Wo + fused (residual = q)
  wmma_gemm<<<dim3(2, 3), 256, 0, stream>>>(aoutb, nullptr, Wo, nullptr, oOb, DF_, DF_, 0);
  k_add<<<(B_SZ * DF_) / 256, 256, 0, stream>>>(oOb, fused, obuf, B_SZ * DF_);

  // hidden = relu(o @ Wd + bd)
  wmma_gemm<<<dim3(2, 3), 256, 0, stream>>>(obuf, nullptr, Wd, bd, hid, DF_, DF_, 3);

  // out = hidden @ W4 + b4
  k_final<<<B_SZ, DF_, 0, stream>>>(hid, W4, b4, out);
}